// GATU_38397007626342
// MI455X (gfx1250) — compile-verified
//
#include <hip/hip_runtime.h>
#include <math.h>

// Problem constants (match reference)
#define NN      50000        // nodes
#define F_IN    128
#define HCH     4            // heads
#define CCH     64           // channels per head
#define FEAT    256          // H*C
#define GG      256          // graphs
#define NEG_SLOPE 0.2f
#define BN_EPS  1e-5f

typedef __attribute__((ext_vector_type(2))) float v2f;
typedef __attribute__((ext_vector_type(8))) float v8f;

// ---------------------------------------------------------------- utilities
__global__ void fill_kernel(float* __restrict__ p, float v, long n) {
    long i = (long)blockIdx.x * blockDim.x + threadIdx.x;
    if (i < n) p[i] = v;
}

__device__ __forceinline__ float atomicMaxFloat(float* addr, float value) {
    // standard bit-punning trick; correct for all non-NaN floats incl. -inf init
    if (value >= 0.0f)
        return __int_as_float(atomicMax((int*)addr, __float_as_int(value)));
    return __uint_as_float(atomicMin((unsigned int*)addr, __float_as_uint(value)));
}

__device__ __forceinline__ float lrelu(float v) {
    return v > 0.0f ? v : NEG_SLOPE * v;
}

// ---------------------------------------------------------------- WMMA GEMM
// C[M,Ncol] = A[M,K] * B[K,Ncol], f32.
// One wave computes a 16x64 strip: 4 x (16x16) accumulators so each A
// fragment feeds 4 v_wmma ops (A traffic /4, wmma:vmem ratio x4).
// Requires M%16==0, K%4==0, Ncol%64==0 (true here: M=50000, K=128/256, Ncol=256).
__global__ void gemm_wmma_f32(const float* __restrict__ A, const float* __restrict__ B,
                              float* __restrict__ C, int M, int K, int Ncol) {
    int wave   = (int)((blockIdx.x * blockDim.x + threadIdx.x) >> 5);
    int lane   = threadIdx.x & 31;
    int groups = Ncol >> 6;             // number of 64-wide column strips
    int tm = wave / groups;
    int tg = wave - tm * groups;
    if (tm * 16 >= M) return;           // uniform across wave

    int half = lane >> 4;               // 0: lanes 0-15, 1: lanes 16-31
    int l    = lane & 15;
    int row  = tm * 16 + l;             // A row held by this lane (both halves)
    int col0 = tg * 64 + l;             // first B/D column held by this lane

    const float* Arow  = A + (size_t)row * K + 2 * half;   // K offset {0,2} per half
    const float* Bbase = B + (size_t)(2 * half) * Ncol + col0;

    v8f acc0 = {}, acc1 = {}, acc2 = {}, acc3 = {};
#if __has_builtin(__builtin_amdgcn_wmma_f32_16x16x4_f32)
    #pragma unroll 2
    for (int k0 = 0; k0 < K; k0 += 4) {
        // A 16x4 layout: vgpr0 = K{0|2}, vgpr1 = K{1|3} (half selects +2)
        float2 av = *(const float2*)(Arow + k0);
        v2f a; a.x = av.x; a.y = av.y;
        // B 4x16 layout per tile: vgpr0 = row K{0|2}, vgpr1 = row K{1|3}, lanes = N
        const float* Br0 = Bbase + (size_t)k0 * Ncol;
        const float* Br1 = Br0 + Ncol;
        v2f b0, b1, b2, b3;
        b0.x = Br0[0];  b0.y = Br1[0];
        b1.x = Br0[16]; b1.y = Br1[16];
        b2.x = Br0[32]; b2.y = Br1[32];
        b3.x = Br0[48]; b3.y = Br1[48];
        acc0 = __builtin_amdgcn_wmma_f32_16x16x4_f32(false, a, false, b0, (short)0, acc0, false, false);
        acc1 = __builtin_amdgcn_wmma_f32_16x16x4_f32(false, a, false, b1, (short)0, acc1, false, false);
        acc2 = __builtin_amdgcn_wmma_f32_16x16x4_f32(false, a, false, b2, (short)0, acc2, false, false);
        acc3 = __builtin_amdgcn_wmma_f32_16x16x4_f32(false, a, false, b3, (short)0, acc3, false, false);
    }
#else
    // scalar fallback (should not be taken on gfx1250)
    for (int k0 = 0; k0 < K; k0 += 4) {
        float2 av = *(const float2*)(Arow + k0);
        acc0[0] += av.x * Bbase[(size_t)k0 * Ncol];
        acc0[0] += av.y * Bbase[(size_t)(k0 + 1) * Ncol];
    }
#endif
    // D 16x16 f32 layout: vgpr r -> row (r + 8*half), lanes = N
    #pragma unroll
    for (int r = 0; r < 8; r++) {
        size_t o = (size_t)(tm * 16 + r + 8 * half) * Ncol + col0;
        C[o]      = acc0[r];
        C[o + 16] = acc1[r];
        C[o + 32] = acc2[r];
        C[o + 48] = acc3[r];
    }
}

// ---------------------------------------------------------------- attention logits
// al_src[n,h] = sum_c h[n,h*64+c]*att_src[h,c]  (same for dst)
__global__ void attn_logits_kernel(const float* __restrict__ h,
                                   const float* __restrict__ asrc,
                                   const float* __restrict__ adst,
                                   float* __restrict__ al_src,
                                   float* __restrict__ al_dst) {
    int idx = blockIdx.x * blockDim.x + threadIdx.x;     // over NN*H
    if (idx >= NN * HCH) return;
    int n  = idx >> 2;
    int hh = idx & 3;
    const float4* hp = (const float4*)(h + (size_t)n * FEAT + hh * CCH);
    const float4* sp = (const float4*)(asrc + hh * CCH);
    const float4* dp = (const float4*)(adst + hh * CCH);
    float ss = 0.0f, sd = 0.0f;
    #pragma unroll 4
    for (int i = 0; i < 16; i++) {
        float4 hv = hp[i], sv = sp[i], dv = dp[i];
        ss += hv.x * sv.x + hv.y * sv.y + hv.z * sv.z + hv.w * sv.w;
        sd += hv.x * dv.x + hv.y * dv.y + hv.z * dv.z + hv.w * dv.w;
    }
    al_src[idx] = ss;
    al_dst[idx] = sd;
}

// ---------------------------------------------------------------- edge passes
__device__ __forceinline__ void edge_nodes(const int* __restrict__ srcI,
                                           const int* __restrict__ dstI,
                                           int e, int nE, int& s, int& d) {
    if (e < nE) { s = srcI[e]; d = dstI[e]; }
    else        { s = d = e - nE; }           // self loops appended
}

__global__ void edge_max_kernel(const int* __restrict__ srcI, const int* __restrict__ dstI,
                                int nE, const float* __restrict__ al_src,
                                const float* __restrict__ al_dst, float* __restrict__ m) {
    int e = blockIdx.x * blockDim.x + threadIdx.x;
    if (e >= nE + NN) return;
    int s, d; edge_nodes(srcI, dstI, e, nE, s, d);
    #pragma unroll
    for (int hh = 0; hh < HCH; hh++) {
        float v = lrelu(al_src[s * HCH + hh] + al_dst[d * HCH + hh]);
        atomicMaxFloat(&m[d * HCH + hh], v);
    }
}

__global__ void edge_denom_kernel(const int* __restrict__ srcI, const int* __restrict__ dstI,
                                  int nE, const float* __restrict__ al_src,
                                  const float* __restrict__ al_dst,
                                  const float* __restrict__ m, float* __restrict__ denom) {
    int e = blockIdx.x * blockDim.x + threadIdx.x;
    if (e >= nE + NN) return;
    int s, d; edge_nodes(srcI, dstI, e, nE, s, d);
    #pragma unroll
    for (int hh = 0; hh < HCH; hh++) {
        float v  = lrelu(al_src[s * HCH + hh] + al_dst[d * HCH + hh]);
        float ex = expf(v - m[d * HCH + hh]);
        atomicAdd(&denom[d * HCH + hh], ex);
    }
}

// one block (256 threads) per edge; thread t -> head t/64, channel t%64
__global__ void edge_agg_kernel(const int* __restrict__ srcI, const int* __restrict__ dstI,
                                int nE, const float* __restrict__ al_src,
                                const float* __restrict__ al_dst,
                                const float* __restrict__ m, const float* __restrict__ denom,
                                const float* __restrict__ h, float* __restrict__ out) {
    int e = blockIdx.x;
    int t = threadIdx.x;               // 0..255
    int hh = t >> 6;
    int s, d; edge_nodes(srcI, dstI, e, nE, s, d);
    float v     = lrelu(al_src[s * HCH + hh] + al_dst[d * HCH + hh]);
    float alpha = expf(v - m[d * HCH + hh]) / (denom[d * HCH + hh] + 1e-16f);
    atomicAdd(&out[(size_t)d * FEAT + t], alpha * h[(size_t)s * FEAT + t]);
}

// ---------------------------------------------------------------- batchnorm
// thread t == column t; coalesced row-major reads; one atomicAdd per block/col
__global__ void bn_stats_kernel(const float* __restrict__ x, const float* __restrict__ b,
                                float* __restrict__ sums, float* __restrict__ sumsq) {
    int c = threadIdx.x;
    float s = 0.0f, s2 = 0.0f;
    for (int n = blockIdx.x; n < NN; n += gridDim.x) {
        float v = x[(size_t)n * FEAT + c] + b[c];
        s += v; s2 += v * v;
    }
    atomicAdd(&sums[c], s);
    atomicAdd(&sumsq[c], s2);
}

// in-place: x = relu(gamma*(x+b-mu)/sqrt(var+eps)+beta); blockIdx = row, tid = col
__global__ void bn_apply_relu_kernel(float* __restrict__ x, const float* __restrict__ b,
                                     const float* __restrict__ gamma, const float* __restrict__ beta,
                                     const float* __restrict__ sums, const float* __restrict__ sumsq) {
    int c = threadIdx.x;
    size_t i = (size_t)blockIdx.x * FEAT + c;
    const float invN = 1.0f / (float)NN;
    float mu  = sums[c] * invN;
    float var = sumsq[c] * invN - mu * mu;
    float v   = x[i] + b[c];
    float y   = gamma[c] * (v - mu) / sqrtf(var + BN_EPS) + beta[c];
    x[i] = fmaxf(y, 0.0f);
}

// in-place: x = relu(x + b); blockIdx = row, tid = col
__global__ void bias_relu_kernel(float* __restrict__ x, const float* __restrict__ b) {
    int c = threadIdx.x;
    size_t i = (size_t)blockIdx.x * FEAT + c;
    x[i] = fmaxf(x[i] + b[c], 0.0f);
}

// ---------------------------------------------------------------- pooling
// blockDim = 512; thread t -> concat channel t (x1 if t<256 else x2)
__global__ void pool_kernel(const float* __restrict__ x1, const float* __restrict__ x2,
                            const int* __restrict__ batch, float* __restrict__ psum,
                            float* __restrict__ pmax, float* __restrict__ pcnt) {
    int t = threadIdx.x;
    for (int n = blockIdx.x; n < NN; n += gridDim.x) {
        int g = batch[n];
        float v = (t < FEAT) ? x1[(size_t)n * FEAT + t]
                             : x2[(size_t)n * FEAT + (t - FEAT)];
        atomicAdd(&psum[(size_t)g * 512 + t], v);
        atomicMaxFloat(&pmax[(size_t)g * 512 + t], v);
        if (t == 0) atomicAdd(&pcnt[g], 1.0f);
    }
}

__global__ void finalize_kernel(const float* __restrict__ psum, const float* __restrict__ pmax,
                                const float* __restrict__ pcnt, float* __restrict__ out) {
    int idx = blockIdx.x * blockDim.x + threadIdx.x;    // over GG*1024
    if (idx >= GG * 1024) return;
    int g = idx >> 10;
    int c = idx & 1023;
    if (c < 512) {
        out[idx] = psum[(size_t)g * 512 + c] / fmaxf(pcnt[g], 1.0f);
    } else {
        float mv = pmax[(size_t)g * 512 + (c - 512)];
        out[idx] = (mv == -INFINITY) ? 0.0f : mv;
    }
}

// ---------------------------------------------------------------- launch
static inline void fill_f(float* p, float v, long n, hipStream_t s) {
    long blocks = (n + 255) / 256;
    fill_kernel<<<(unsigned)blocks, 256, 0, s>>>(p, v, n);
}

extern "C" void kernel_launch(void* const* d_in, const int* in_sizes, int n_in,
                              void* d_out, int out_size, void* d_ws, size_t ws_size,
                              hipStream_t stream) {
    // inputs (setup_inputs dict order)
    const float* x        = (const float*)d_in[0];
    const int*   ei       = (const int*)d_in[1];      // [2,E] flat: src then dst
    const int*   batch    = (const int*)d_in[2];
    const float* W1       = (const float*)d_in[3];
    const float* att_src1 = (const float*)d_in[4];
    const float* att_dst1 = (const float*)d_in[5];
    const float* b1       = (const float*)d_in[6];
    const float* gamma    = (const float*)d_in[7];
    const float* beta     = (const float*)d_in[8];
    const float* W2       = (const float*)d_in[9];
    const float* att_src2 = (const float*)d_in[10];
    const float* att_dst2 = (const float*)d_in[11];
    const float* b2       = (const float*)d_in[12];
    float* out = (float*)d_out;

    const int E  = in_sizes[1] / 2;
    const int EE = E + NN;                    // edges + self loops
    const int* srcI = ei;
    const int* dstI = ei + E;

    // workspace layout (floats)
    const long NF = (long)NN * FEAT;          // 12.8M
    const long NH = (long)NN * HCH;           // 200k
    const long GP = (long)GG * 512;           // 131072
    float* ws    = (float*)d_ws;
    float* hbuf  = ws;            // [N,256]  h1 then h2
    float* x1    = hbuf + NF;     // [N,256]  layer1 out -> BN+relu in place
    float* x2    = x1 + NF;       // [N,256]  layer2 out -> bias+relu in place
    float* alS   = x2 + NF;       // [N,4]
    float* alD   = alS + NH;      // [N,4]
    float* mbuf  = alD + NH;      // [N,4]
    float* dbuf  = mbuf + NH;     // [N,4]
    float* sums  = dbuf + NH;     // [256]
    float* sumsq = sums + FEAT;   // [256]
    float* psum  = sumsq + FEAT;  // [G,512]
    float* pmax  = psum + GP;     // [G,512]
    float* pcnt  = pmax + GP;     // [G]

    const int edgeBlocks = (EE + 255) / 256;
    const int attnBlocks = (NN * HCH + 255) / 256;
    // GEMM: one wave per 16x64 strip -> (M/16)*(Ncol/64) waves, 4 waves/block
    const int gemmBlocks = (NN / 16) * (FEAT / 64) / 4;   // 3125

    // ---------------- layer 1 ----------------
    fill_f(x1, 0.0f, NF, stream);
    fill_f(mbuf, -INFINITY, NH, stream);
    fill_f(dbuf, 0.0f, NH, stream);
    fill_f(sums, 0.0f, 2 * FEAT, stream);

    gemm_wmma_f32<<<gemmBlocks, 128, 0, stream>>>(x, W1, hbuf, NN, F_IN, FEAT);
    attn_logits_kernel<<<attnBlocks, 256, 0, stream>>>(hbuf, att_src1, att_dst1, alS, alD);
    edge_max_kernel<<<edgeBlocks, 256, 0, stream>>>(srcI, dstI, E, alS, alD, mbuf);
    edge_denom_kernel<<<edgeBlocks, 256, 0, stream>>>(srcI, dstI, E, alS, alD, mbuf, dbuf);
    edge_agg_kernel<<<EE, 256, 0, stream>>>(srcI, dstI, E, alS, alD, mbuf, dbuf, hbuf, x1);

    bn_stats_kernel<<<256, 256, 0, stream>>>(x1, b1, sums, sumsq);
    bn_apply_relu_kernel<<<NN, 256, 0, stream>>>(x1, b1, gamma, beta, sums, sumsq);

    // ---------------- layer 2 ----------------
    fill_f(x2, 0.0f, NF, stream);
    fill_f(mbuf, -INFINITY, NH, stream);
    fill_f(dbuf, 0.0f, NH, stream);

    gemm_wmma_f32<<<gemmBlocks, 128, 0, stream>>>(x1, W2, hbuf, NN, FEAT, FEAT);
    attn_logits_kernel<<<attnBlocks, 256, 0, stream>>>(hbuf, att_src2, att_dst2, alS, alD);
    edge_max_kernel<<<edgeBlocks, 256, 0, stream>>>(srcI, dstI, E, alS, alD, mbuf);
    edge_denom_kernel<<<edgeBlocks, 256, 0, stream>>>(srcI, dstI, E, alS, alD, mbuf, dbuf);
    edge_agg_kernel<<<EE, 256, 0, stream>>>(srcI, dstI, E, alS, alD, mbuf, dbuf, hbuf, x2);
    bias_relu_kernel<<<NN, 256, 0, stream>>>(x2, b2);

    // ---------------- pooling ----------------
    fill_f(psum, 0.0f, GP, stream);
    fill_f(pmax, -INFINITY, GP, stream);
    fill_f(pcnt, 0.0f, GG, stream);
    pool_kernel<<<1024, 512, 0, stream>>>(x1, x2, batch, psum, pmax, pcnt);
    finalize_kernel<<<(GG * 1024 + 255) / 256, 256, 0, stream>>>(psum, pmax, pcnt, out);
}